// model_BoolQA_53094385713544
// MI455X (gfx1250) — compile-verified
//
#include <hip/hip_runtime.h>
#include <hip/hip_bf16.h>

// Problem constants (match reference)
#define BB 128
#define SS 512
#define HH 768

typedef float  f4  __attribute__((ext_vector_type(4)));
typedef float  v2f __attribute__((ext_vector_type(2)));
typedef float  v8f __attribute__((ext_vector_type(8)));

// ---------------------------------------------------------------------------
// Kernel 1: deterministic split-span pooling partial sums.
// grid = (B, R); block = 192 threads; thread t owns float4 at column 4*t.
// Block (b, r) sums rows s+1+r, s+1+r+R, ... (< e) and writes its partial
// row of H floats to ws[(b*R + r)*H ..]. Always writes (zeros if no rows),
// so the workspace is fully re-initialized every call (determinism, and no
// float atomics -> bitwise-identical across graph replays).
// ---------------------------------------------------------------------------
template <int R>
__global__ __launch_bounds__(192)
void pool_partial_kernel(const float* __restrict__ X,
                         const int* __restrict__ sep_s,
                         const int* __restrict__ sep_e,
                         float* __restrict__ partial)
{
    const int b = blockIdx.x;
    const int r = blockIdx.y;
    const int t = threadIdx.x;                 // 0..191, one float4 each
    const int s = sep_s[b];
    const int e = sep_e[b];

    const f4* base = (const f4*)(X + (size_t)b * SS * HH) + t;
    const int rowStride = HH / 4;              // 192 float4 per row

    f4 acc0 = {0.f, 0.f, 0.f, 0.f};
    f4 acc1 = {0.f, 0.f, 0.f, 0.f};

    int row = s + 1 + r;
    // 2x unrolled: two independent non-temporal b128 loads in flight per thread
    for (; row + R < e; row += 2 * R) {
        acc0 += __builtin_nontemporal_load(base + (size_t)row * rowStride);
        acc1 += __builtin_nontemporal_load(base + (size_t)(row + R) * rowStride);
    }
    if (row < e) {
        acc0 += __builtin_nontemporal_load(base + (size_t)row * rowStride);
    }
    f4 acc = acc0 + acc1;

    f4* pp = (f4*)(partial + ((size_t)b * R + r) * HH);
    pp[t] = acc;
}

// ---------------------------------------------------------------------------
// Kernel 2: out[128,2] = (sum_partials / count) @ W^T + bias via
// V_WMMA_F32_16X16X4_F32. grid = 8 blocks (M-tiles of 16), 32 threads (1 wave).
// K loop: 192 steps of 16x16x4. N is padded 2 -> 16 with zero B columns.
//
// A 16x4 f32 layout: lanes 0-15 hold M=0..15 with {K=0,K=1} in the 2 VGPRs,
// lanes 16-31 hold M=0..15 with {K=2,K=3}. B 4x16 mirrors that in K.
// C/D 16x16 f32: VGPR v = row (v) for lanes 0-15, row (v+8) for lanes 16-31,
// N = lane & 15.
// ---------------------------------------------------------------------------
template <int R>
__global__ __launch_bounds__(32)
void wmma_project_kernel(const float* __restrict__ partial,
                         const int* __restrict__ sep_s,
                         const int* __restrict__ sep_e,
                         const float* __restrict__ W,
                         const float* __restrict__ bias,
                         float* __restrict__ out)
{
    const int lane   = threadIdx.x;        // 0..31
    const int m_base = blockIdx.x * 16;
    const int half   = lane >> 4;          // 0: K=0,1   1: K=2,3
    const int n      = lane & 15;          // N index (valid cols: 0,1)
    const int koff   = half * 2;

    const int    rowA = m_base + n;        // this lane's A-matrix row
    const float* prow = partial + (size_t)rowA * R * HH + koff;

    // Clamped, always-in-bounds W row pointer; lanes n>=2 load row 0 but the
    // value is masked to zero below (branch-free, no OOB address formed).
    const float* wrow = W + (size_t)((n < 2) ? n : 0) * HH + koff;
    const float  wmask = (n < 2) ? 1.f : 0.f;

    v8f acc = {0.f, 0.f, 0.f, 0.f, 0.f, 0.f, 0.f, 0.f};

    #pragma unroll 2
    for (int k = 0; k < HH; k += 4) {
        // A element pair: combine the R deterministic partial sums
        // (R is a compile-time constant -> straight-line b64 loads)
        float ax = 0.f, ay = 0.f;
        #pragma unroll
        for (int j = 0; j < R; ++j) {
            const v2f p = *(const v2f*)(prow + (size_t)j * HH + k);
            ax += p.x;
            ay += p.y;
        }
        v2f a = {ax, ay};

        // B element pair: W^T padded to 16 columns (cols >= 2 zeroed by mask)
        const v2f w = *(const v2f*)(wrow + k);
        v2f bmat = {w.x * wmask, w.y * wmask};

        // (neg_a, A, neg_b, B, c_mod, C, reuse_a, reuse_b)
        acc = __builtin_amdgcn_wmma_f32_16x16x4_f32(
            false, a, false, bmat, (short)0, acc, false, false);
    }

    // Epilogue: scale by 1/count per output row, add bias, store cols 0..1.
    // (All WMMAs are done before this divergent branch -> EXEC all-1s rule OK.)
    if (n < 2) {
        const float bn = bias[n];
        #pragma unroll
        for (int v = 0; v < 8; ++v) {
            const int m = m_base + v + half * 8;
            const float cnt = (float)(sep_e[m] - sep_s[m] - 1); // >= 1 by setup
            out[m * 2 + n] = acc[v] / cnt + bn;
        }
    }
}

// ---------------------------------------------------------------------------
// Launch
// ---------------------------------------------------------------------------
extern "C" void kernel_launch(void* const* d_in, const int* in_sizes, int n_in,
                              void* d_out, int out_size, void* d_ws, size_t ws_size,
                              hipStream_t stream)
{
    const float* X     = (const float*)d_in[0];   // [128,512,768] f32
    const int*   seps  = (const int*)  d_in[1];   // [128] i32
    const int*   sepe  = (const int*)  d_in[2];   // [128] i32
    const float* W     = (const float*)d_in[3];   // [2,768] f32
    const float* bias  = (const float*)d_in[4];   // [2] f32
    float*       out   = (float*)d_out;           // [128,2] f32
    float*       ws    = (float*)d_ws;

    // Split factor for the pooling reduction (deterministic, no atomics).
    const size_t need8 = (size_t)BB * 8 * HH * sizeof(float);   // 3 MB
    if (ws_size >= need8) {
        constexpr int R = 8;
        dim3 grid1(BB, R, 1);
        pool_partial_kernel<R><<<grid1, 192, 0, stream>>>(X, seps, sepe, ws);
        wmma_project_kernel<R><<<BB / 16, 32, 0, stream>>>(ws, seps, sepe, W, bias, out);
    } else {
        constexpr int R = 1;
        dim3 grid1(BB, R, 1);
        pool_partial_kernel<R><<<grid1, 192, 0, stream>>>(X, seps, sepe, ws);
        wmma_project_kernel<R><<<BB / 16, 32, 0, stream>>>(ws, seps, sepe, W, bias, out);
    }
}